// TropicalDotProduct_9251359556268
// MI455X (gfx1250) — compile-verified
//
#include <hip/hip_runtime.h>
#include <stdint.h>

// Tropical (max-plus) "GEMM": out[n,m] = max_k(|x[n,k]| + |w[m,k]|)
// N = M = 8192, K = 128, fp32 in / fp32 out.
//
// MI455X reasoning:
//  * No WMMA path exists for a (max,+) semiring -> VALU kernel by necessity.
//  * Inputs are 8.4 MB total (resident in 192 MB L2); output stream is 256 MB
//    (~11.5 us at 23.3 TB/s). Compute is 2*N*M*K = 17.2 G VALU ops, ~10x the
//    store time -> VALU bound. 8x8 register tile/thread, |.| folded into VOP3
//    abs source modifiers, add/max pairs dual-issued as VOPD on wave32.
//  * CDNA5-specific data movement: per-lane global_load_async_to_lds_b32 does
//    the global->LDS *transpose* scatter with no VGPR staging (ASYNCcnt), then
//    s_wait_asynccnt + workgroup barrier.
//  * Output stores are non-temporal (b128 + NT hint): the 256 MB write stream
//    must not evict the small input working set from L2.

#define TILE       128
#define LDS_STRIDE 132   // 128 + 4 pad: keeps ds_load_b128 16B-aligned, spreads banks

typedef __attribute__((ext_vector_type(4))) float v4f;   // clang vector type for NT stores

__global__ __launch_bounds__(256, 2)   // 132 KB LDS -> exactly 2 workgroups / 320 KB WGP
void tropical_maxplus_kernel(const float* __restrict__ x,
                             const float* __restrict__ w,
                             float* __restrict__ out,
                             int M /* = 8192, out row stride */) {
    // Transposed tiles: l?s[k * LDS_STRIDE + row], k = 0..127, row = 0..127
    __shared__ float lxs[TILE * LDS_STRIDE];   // 67,584 B
    __shared__ float lws[TILE * LDS_STRIDE];   // 67,584 B  (total ~132 KB)

    const int t  = threadIdx.x;      // 0..255
    const int tx = t & 15;           // output m sub-block
    const int ty = t >> 4;           // output n sub-block
    const int n0 = blockIdx.y * TILE;
    const int m0 = blockIdx.x * TILE;

    // ---------------- async fill: global [row][k] -> LDS transposed [k][row] ----------
    // Each thread owns one fixed k-column and copies 64 rows (stride 2) per tile.
    {
        const int k    = t & 127;    // fixed k for this thread
        const int rbeg = t >> 7;     // 0 or 1
        uint64_t gx = (uint64_t)(uintptr_t)(x + (uint64_t)(n0 + rbeg) * 128u + (unsigned)k);
        uint64_t gw = (uint64_t)(uintptr_t)(w + (uint64_t)(m0 + rbeg) * 128u + (unsigned)k);
        // Flat shared pointers: low 32 bits are the LDS byte offset (LDS aperture).
        uint32_t lx = (uint32_t)(uintptr_t)(&lxs[k * LDS_STRIDE + rbeg]);
        uint32_t lw = (uint32_t)(uintptr_t)(&lws[k * LDS_STRIDE + rbeg]);
#pragma unroll 4
        for (int i = 0; i < 64; ++i) {
            asm volatile("global_load_async_to_lds_b32 %0, %1, off"
                         :: "v"(lx), "v"(gx) : "memory");
            asm volatile("global_load_async_to_lds_b32 %0, %1, off"
                         :: "v"(lw), "v"(gw) : "memory");
            lx += 8u;      // row += 2  -> 2 floats in transposed layout
            lw += 8u;
            gx += 1024ull; // 2 rows * 128 k * 4 B
            gw += 1024ull;
        }
    }
    asm volatile("s_wait_asynccnt 0x0" ::: "memory");
    __syncthreads();

    // ---------------- compute: 8x8 register tile per thread --------------------------
    float acc[8][8];
#pragma unroll
    for (int i = 0; i < 8; ++i)
#pragma unroll
        for (int j = 0; j < 8; ++j)
            acc[i][j] = -__builtin_huge_valf();

    const float* ax = &lxs[ty * 8];
    const float* bw = &lws[tx * 8];

#pragma unroll 2
    for (int k = 0; k < TILE; ++k) {
        const int koff = k * LDS_STRIDE;
        const v4f a0 = *reinterpret_cast<const v4f*>(ax + koff);
        const v4f a1 = *reinterpret_cast<const v4f*>(ax + koff + 4);
        const v4f b0 = *reinterpret_cast<const v4f*>(bw + koff);
        const v4f b1 = *reinterpret_cast<const v4f*>(bw + koff + 4);

        // |.| folds into VOP3 abs source modifiers on v_add_f32 (zero extra VALU).
        float a[8] = { __builtin_fabsf(a0.x), __builtin_fabsf(a0.y),
                       __builtin_fabsf(a0.z), __builtin_fabsf(a0.w),
                       __builtin_fabsf(a1.x), __builtin_fabsf(a1.y),
                       __builtin_fabsf(a1.z), __builtin_fabsf(a1.w) };
        float b[8] = { __builtin_fabsf(b0.x), __builtin_fabsf(b0.y),
                       __builtin_fabsf(b0.z), __builtin_fabsf(b0.w),
                       __builtin_fabsf(b1.x), __builtin_fabsf(b1.y),
                       __builtin_fabsf(b1.z), __builtin_fabsf(b1.w) };

#pragma unroll
        for (int i = 0; i < 8; ++i)
#pragma unroll
            for (int j = 0; j < 8; ++j)
                acc[i][j] = fmaxf(acc[i][j], a[i] + b[j]);
    }

    // ---------------- store: 2x non-temporal global_store_b128 per output row --------
    float* orow = out + (uint64_t)(n0 + ty * 8) * (uint64_t)M + (uint64_t)(m0 + tx * 8);
#pragma unroll
    for (int i = 0; i < 8; ++i) {
        v4f lo = { acc[i][0], acc[i][1], acc[i][2], acc[i][3] };
        v4f hi = { acc[i][4], acc[i][5], acc[i][6], acc[i][7] };
        __builtin_nontemporal_store(lo, reinterpret_cast<v4f*>(orow));
        __builtin_nontemporal_store(hi, reinterpret_cast<v4f*>(orow + 4));
        orow += M;
    }
}

extern "C" void kernel_launch(void* const* d_in, const int* in_sizes, int n_in,
                              void* d_out, int out_size, void* d_ws, size_t ws_size,
                              hipStream_t stream) {
    (void)n_in; (void)out_size; (void)d_ws; (void)ws_size;
    const float* x = (const float*)d_in[0];   // [N, 128]
    const float* w = (const float*)d_in[1];   // [M, 128]
    float* out = (float*)d_out;               // [N, M]

    const int K = 128;
    const int N = in_sizes[0] / K;            // 8192
    const int M = in_sizes[1] / K;            // 8192

    dim3 grid((unsigned)(M / TILE), (unsigned)(N / TILE));   // 64 x 64
    tropical_maxplus_kernel<<<grid, 256, 0, stream>>>(x, w, out, M);
}